// DynamicIcoConv_14422500180038
// MI455X (gfx1250) — compile-verified
//
#include <hip/hip_runtime.h>

typedef __attribute__((ext_vector_type(2))) float v2f;
typedef __attribute__((ext_vector_type(8))) float v8f;

namespace {
constexpr int kOrder  = 60;
constexpr int kV      = 642;
constexpr int kVH     = 321;
constexpr int kN      = 64;
constexpr int kM      = 64;
constexpr int kK2     = kVH * kN;            // 20544
constexpr int kHTiles = (kVH + 15) / 16;     // 21
constexpr int kOTiles = (kOrder + 15) / 16;  // 4
constexpr int kKSplit = 16;
constexpr int kKChunk = kK2 / kKSplit;       // 1284
constexpr int kKSteps = kKChunk / 4;         // 321
}

// ---------------------------------------------------------------------------
// Stage 1: S[b,o,h,n] = sum_v hqr[o,v,h] * x[b,v,n]
// One wave computes a 16(h) x 64(n) strip via V_WMMA_F32_16X16X4_F32.
// A(16x4): lane<16 -> (M=h, K=kk+{0,1}); lane>=16 -> (M=h, K=kk+{2,3})
// B(4x16): lane<16 -> (K=kk+{0,1}, N=n);  lane>=16 -> (K=kk+{2,3}, N=n)
// Edge handling: A-row padding (h>=VH) uses CLAMPED addresses (garbage rows
// only affect D rows we never store) -> no divergent loads in the hot loop.
// K padding (v>=642, tail only) zeroes B via load-then-select (v_cndmask).
// ---------------------------------------------------------------------------
__global__ __launch_bounds__(32)
void ico_stage1_wmma(const float* __restrict__ x,
                     const float* __restrict__ hqr,
                     float* __restrict__ shifted) {
  const int lane = threadIdx.x & 31;
  const int half = lane >> 4;
  const int l16  = lane & 15;
  const int h0   = blockIdx.x * 16;
  const int o    = blockIdx.y;
  const int b    = blockIdx.z;

  const float* __restrict__ hq = hqr + (size_t)o * kV * kVH; // [v][h]
  const float* __restrict__ xb = x   + (size_t)b * kV * kN;  // [v][n]

  const int  hA  = h0 + l16;
  const bool hok = (hA < kVH);
  const int  hC  = hok ? hA : (kVH - 1);      // clamped A column (safe address)
  const float* __restrict__ ap = hq + hC;     // index with v*kVH

  v8f acc0 = {}, acc1 = {}, acc2 = {}, acc3 = {};

  // 640 of 642 v-values: full, unguarded K=4 steps
#pragma unroll 2
  for (int kk = 0; kk < 640; kk += 4) {
    const int v0 = kk + 2 * half;
    v2f a;
    a.x = ap[(size_t)v0 * kVH];
    a.y = ap[(size_t)(v0 + 1) * kVH];

    const float* r0 = xb + (size_t)v0 * kN;
    const float* r1 = r0 + kN;
    v2f b0, b1, b2, b3;
    b0.x = r0[l16];      b0.y = r1[l16];
    b1.x = r0[16 + l16]; b1.y = r1[16 + l16];
    b2.x = r0[32 + l16]; b2.y = r1[32 + l16];
    b3.x = r0[48 + l16]; b3.y = r1[48 + l16];

    acc0 = __builtin_amdgcn_wmma_f32_16x16x4_f32(false, a, false, b0, (short)0, acc0, false, false);
    acc1 = __builtin_amdgcn_wmma_f32_16x16x4_f32(false, a, false, b1, (short)0, acc1, false, false);
    acc2 = __builtin_amdgcn_wmma_f32_16x16x4_f32(false, a, false, b2, (short)0, acc2, false, false);
    acc3 = __builtin_amdgcn_wmma_f32_16x16x4_f32(false, a, false, b3, (short)0, acc3, false, false);
  }

  // tail step: v = 640,641 valid; 642,643 are K-padding.
  // Load everything from clamped rows (640,641), then zero B for padded K
  // lanes with a select. A may stay garbage: A(M,kpad)*B(kpad,N)=0 since B=0.
  {
    const bool k0ok = (half == 0);            // lanes>=16 carry padded K
    v2f a;
    a.x = ap[(size_t)640 * kVH];
    a.y = ap[(size_t)641 * kVH];

    const float* r0 = xb + (size_t)640 * kN;
    const float* r1 = r0 + kN;
    float t0a = r0[l16],      t0b = r1[l16];
    float t1a = r0[16 + l16], t1b = r1[16 + l16];
    float t2a = r0[32 + l16], t2b = r1[32 + l16];
    float t3a = r0[48 + l16], t3b = r1[48 + l16];
    v2f b0, b1, b2, b3;
    b0.x = k0ok ? t0a : 0.0f; b0.y = k0ok ? t0b : 0.0f;
    b1.x = k0ok ? t1a : 0.0f; b1.y = k0ok ? t1b : 0.0f;
    b2.x = k0ok ? t2a : 0.0f; b2.y = k0ok ? t2b : 0.0f;
    b3.x = k0ok ? t3a : 0.0f; b3.y = k0ok ? t3b : 0.0f;

    acc0 = __builtin_amdgcn_wmma_f32_16x16x4_f32(false, a, false, b0, (short)0, acc0, false, false);
    acc1 = __builtin_amdgcn_wmma_f32_16x16x4_f32(false, a, false, b1, (short)0, acc1, false, false);
    acc2 = __builtin_amdgcn_wmma_f32_16x16x4_f32(false, a, false, b2, (short)0, acc2, false, false);
    acc3 = __builtin_amdgcn_wmma_f32_16x16x4_f32(false, a, false, b3, (short)0, acc3, false, false);
  }

  // D layout: lane<16 VGPR r -> (M=r, N=l16); lane>=16 -> (M=r+8, N=l16)
  float* __restrict__ dst = shifted + ((size_t)b * kOrder + o) * (size_t)kVH * kN;
#pragma unroll
  for (int r = 0; r < 8; ++r) {
    const int h = h0 + r + 8 * half;
    if (h < kVH) {
      float* row = dst + (size_t)h * kN;
      row[l16]      = acc0[r];
      row[16 + l16] = acc1[r];
      row[32 + l16] = acc2[r];
      row[48 + l16] = acc3[r];
    }
  }
}

// ---------------------------------------------------------------------------
// Stage 2: out[b,o,m] += sum_k S[b,o,k] * w[b,k,m]  (k = h*64+n, split-K 16)
// One wave computes a 16(o) x 64(m) partial tile over a 1284-long K chunk,
// accumulated into pre-zeroed d_out with global_atomic_add_f32.
// Padded o rows (o>=60) use clamped A addresses; their results are discarded.
// ---------------------------------------------------------------------------
__global__ __launch_bounds__(32)
void ico_stage2_wmma(const float* __restrict__ shifted,
                     const float* __restrict__ weight,
                     float* __restrict__ out) {
  const int lane = threadIdx.x & 31;
  const int half = lane >> 4;
  const int l16  = lane & 15;
  const int o0   = blockIdx.y * 16;
  const int b    = blockIdx.z;
  const size_t kbase = (size_t)blockIdx.x * kKChunk;

  const float* __restrict__ A = shifted + (size_t)b * kOrder * kK2;      // [o][k]
  const float* __restrict__ W = weight  + (size_t)b * (size_t)kK2 * kM;  // [k][m]

  const int oA = o0 + l16;
  const int oC = (oA < kOrder) ? oA : (kOrder - 1);  // clamped A row
  const float* __restrict__ arow = A + (size_t)oC * kK2 + kbase;

  v8f acc0 = {}, acc1 = {}, acc2 = {}, acc3 = {};

#pragma unroll 2
  for (int s = 0; s < kKSteps; ++s) {
    const size_t kk = (size_t)s * 4 + 2 * half;
    const v2f a = *reinterpret_cast<const v2f*>(arow + kk);  // 8B aligned

    const float* w0 = W + (kbase + kk) * kM;
    const float* w1 = w0 + kM;
    v2f b0, b1, b2, b3;
    b0.x = w0[l16];      b0.y = w1[l16];
    b1.x = w0[16 + l16]; b1.y = w1[16 + l16];
    b2.x = w0[32 + l16]; b2.y = w1[32 + l16];
    b3.x = w0[48 + l16]; b3.y = w1[48 + l16];

    acc0 = __builtin_amdgcn_wmma_f32_16x16x4_f32(false, a, false, b0, (short)0, acc0, false, false);
    acc1 = __builtin_amdgcn_wmma_f32_16x16x4_f32(false, a, false, b1, (short)0, acc1, false, false);
    acc2 = __builtin_amdgcn_wmma_f32_16x16x4_f32(false, a, false, b2, (short)0, acc2, false, false);
    acc3 = __builtin_amdgcn_wmma_f32_16x16x4_f32(false, a, false, b3, (short)0, acc3, false, false);
  }

  float* __restrict__ outb = out + (size_t)b * kOrder * kM;
#pragma unroll
  for (int r = 0; r < 8; ++r) {
    const int o = o0 + r + 8 * half;
    if (o < kOrder) {
      float* row = outb + (size_t)o * kM;
      atomicAdd(row + l16,      acc0[r]);
      atomicAdd(row + 16 + l16, acc1[r]);
      atomicAdd(row + 32 + l16, acc2[r]);
      atomicAdd(row + 48 + l16, acc3[r]);
    }
  }
}

__global__ void ico_zero_f32(float* __restrict__ p, int n) {
  const int i = blockIdx.x * blockDim.x + threadIdx.x;
  if (i < n) p[i] = 0.0f;
}

extern "C" void kernel_launch(void* const* d_in, const int* in_sizes, int n_in,
                              void* d_out, int out_size, void* d_ws, size_t ws_size,
                              hipStream_t stream) {
  const float* x   = (const float*)d_in[0];  // (8, 642, 64)
  const float* w   = (const float*)d_in[1];  // (8, 321, 64, 64)
  const float* hqr = (const float*)d_in[2];  // (60, 642, 321)
  float* out = (float*)d_out;                // (8, 60, 64)
  float* shifted = (float*)d_ws;             // (8, 60, 321, 64) f32 = 37.6 MB

  ico_zero_f32<<<(out_size + 255) / 256, 256, 0, stream>>>(out, out_size);
  ico_stage1_wmma<<<dim3(kHTiles, kOrder, 8), 32, 0, stream>>>(x, hqr, shifted);
  ico_stage2_wmma<<<dim3(kKSplit, kOTiles, 8), 32, 0, stream>>>(shifted, w, out);
}